// torch_env_renderer_59416577573397
// MI455X (gfx1250) — compile-verified
//
#include <hip/hip_runtime.h>

// ---------------------------------------------------------------------------
// Env-map diffuse renderer for gfx1250 (MI455X, wave32, WMMA).
//
//   dots  = relu( Matn(N,3) x Matl(3,M) )      -> V_WMMA_F32_16X16X4_F32
//   light = dots(N,M) x hdr(M,3)               -> per-lane FMAs from LDS
//   out   = light/50 * albedo
//
// Per wave: 16 pixels, loop over 32 tiles of 16 lights -> 32 WMMAs / wave.
// relu is a single v_med3_f32 (clamp to [0, FLT_MAX]) since WMMA outputs
// are finite here.
// ---------------------------------------------------------------------------

typedef float v2f __attribute__((ext_vector_type(2)));
typedef float v8f __attribute__((ext_vector_type(8)));

#define HDR_M 512      // 16*32 light directions
#define NPIX  262144   // 512*512 pixels per batch
#define PI_F  3.14159265358979323846f
#define BIG_F 3.0e38f

__global__ __launch_bounds__(256) void env_render_wmma(
    const float* __restrict__ env,     // (B,16,32,3)
    const float* __restrict__ normal,  // (B,512,512,3)
    const float* __restrict__ albedo,  // (B,3,512,512)
    float* __restrict__ out)           // (B,3,512,512)
{
  __shared__ float hdr_lds[HDR_M * 3];  // 6 KB : env * sin(phi_row)
  __shared__ float a_lds[32 * 32 * 2];  // 8 KB : WMMA A-fragments per (tile,lane)

  const int tid  = threadIdx.x;
  const int lane = tid & 31;
  const int wave = tid >> 5;
  const int b        = blockIdx.x >> 11;   // 2048 blocks per batch image
  const int pixBlock = blockIdx.x & 2047;

  // ---- build hdr = env * sin(phi) in LDS (1536 floats, 6 per thread) ------
  #pragma unroll
  for (int k = 0; k < 6; ++k) {
    int i   = tid + k * 256;           // flat index m*3+c
    int row = i / 96;                  // m / 32  (phi row)
    float sp = __sinf((float)row * (PI_F / 16.0f));
    hdr_lds[i] = env[(size_t)b * (HDR_M * 3) + i] * sp;
  }

  // ---- build A-fragments (light dirs in 16x4 f32 A register layout) -------
  // lanes 0-15 hold K=0,1 ; lanes 16-31 hold K=2,3 (K=3 is zero pad)
  #pragma unroll
  for (int k = 0; k < 4; ++k) {
    int f = tid + k * 256;             // 0..1023 = (tile t, lane L)
    int t = f >> 5, L = f & 31;
    int m = t * 16 + (L & 15);         // light row index for this A element
    int row = m >> 5, col = m & 31;
    float phi   = (float)row * (PI_F / 16.0f);
    float theta = (float)col * (PI_F / 16.0f);   // 2*pi/32 * col
    float spv = __sinf(phi),  cpv = __cosf(phi);
    float stv = __sinf(theta), ctv = __cosf(theta);
    float dx = stv * spv, dy = cpv, dz = -ctv * spv;
    float inv = rsqrtf(dx * dx + dy * dy + dz * dz);
    dx *= inv; dy *= inv; dz *= inv;
    a_lds[2 * f + 0] = (L < 16) ? dx : dz;       // K0 / K2
    a_lds[2 * f + 1] = (L < 16) ? dy : 0.0f;     // K1 / K3(pad)
  }
  __syncthreads();

  // ---- B fragment: decoded, channel-flipped normal of this wave's 16 pixels
  const int p = pixBlock * 128 + wave * 16 + (lane & 15);
  const size_t nbase = ((size_t)b * NPIX + p) * 3;
  float n0 = normal[nbase + 0];
  float n1 = normal[nbase + 1];
  float n2 = normal[nbase + 2];
  v2f Bv;
  Bv.x = (lane < 16) ? (2.0f * (n2 - 0.5f)) : (2.0f * (n0 - 0.5f)); // K0 / K2
  Bv.y = (lane < 16) ? (2.0f * (n1 - 0.5f)) : 0.0f;                 // K1 / K3

  float acc0 = 0.0f, acc1 = 0.0f, acc2 = 0.0f;
  const int half8 = (lane < 16) ? 0 : 8;   // D rows: VGPR r -> light r (+8)

  for (int t = 0; t < 32; ++t) {
    v2f Av;
    Av.x = a_lds[(t * 32 + lane) * 2 + 0];
    Av.y = a_lds[(t * 32 + lane) * 2 + 1];
    v8f c = {};
    // D(16x16) = A(16 lights x 4) * B(4 x 16 pixels) + 0
    v8f d = __builtin_amdgcn_wmma_f32_16x16x4_f32(
        /*neg_a=*/false, Av, /*neg_b=*/false, Bv,
        /*c_mod=*/(short)0, c, /*reuse_a=*/false, /*reuse_b=*/false);

    #pragma unroll
    for (int r = 0; r < 8; ++r) {
      // relu as a single v_med3_f32: median(0, x, BIG) == clamp(x, 0, BIG)
      float dr = __builtin_amdgcn_fmed3f(0.0f, d[r], BIG_F);
      const float* w = &hdr_lds[(t * 16 + r + half8) * 3];
      acc0 = fmaf(dr, w[0], acc0);
      acc1 = fmaf(dr, w[1], acc1);
      acc2 = fmaf(dr, w[2], acc2);
    }
  }

  // fold the two half-wave light partitions (rows 0-7 vs 8-15 of each tile)
  acc0 += __shfl_xor(acc0, 16, 32);
  acc1 += __shfl_xor(acc1, 16, 32);
  acc2 += __shfl_xor(acc2, 16, 32);

  if (lane < 16) {
    const size_t base = (size_t)b * 3 * NPIX + p;
    const float s = 1.0f / 50.0f;
    out[base           ] = acc0 * s * albedo[base           ];
    out[base +     NPIX] = acc1 * s * albedo[base +     NPIX];
    out[base + 2 * NPIX] = acc2 * s * albedo[base + 2 * NPIX];
  }
}

extern "C" void kernel_launch(void* const* d_in, const int* in_sizes, int n_in,
                              void* d_out, int out_size, void* d_ws, size_t ws_size,
                              hipStream_t stream) {
  (void)n_in; (void)out_size; (void)d_ws; (void)ws_size;
  const float* env    = (const float*)d_in[0];
  const float* normal = (const float*)d_in[1];
  const float* albedo = (const float*)d_in[2];
  float* out = (float*)d_out;

  const int B = in_sizes[0] / (16 * 32 * 3);   // batch from env size
  dim3 grid(B * 2048), block(256);             // 128 pixels per block (8 waves)
  hipLaunchKernelGGL(env_render_wmma, grid, block, 0, stream,
                     env, normal, albedo, out);
}